// GIN_82068235092723
// MI455X (gfx1250) — compile-verified
//
#include <hip/hip_runtime.h>

#define N_NODES 100000
#define N_EDGES 3200000
#define KDIM    128   // input dim of every layer (128 -> ... -> 128 -> 64)

typedef float v2f __attribute__((ext_vector_type(2)));
typedef float v8f __attribute__((ext_vector_type(8)));

// ---------------------------------------------------------------------------
// agg[v] = h[v]   (seed accumulator with h so scatter produces X = h + sum)
// ---------------------------------------------------------------------------
__global__ void __launch_bounds__(256)
gin_copy_f4(float4* __restrict__ dst, const float4* __restrict__ src, int n4) {
    int t = blockIdx.x * 256 + threadIdx.x;
    if (t < n4) dst[t] = src[t];
}

// ---------------------------------------------------------------------------
// Scatter-add: one wave handles 4 consecutive edges. Edge indices are made
// wave-uniform via readfirstlane so they take the scalar (s_load) path; each
// lane then does one b128 gather of the 128-float source row slice and four
// no-return global_atomic_add_f32 into the destination row. All rows are
// L2-resident (51 MB each).
// ---------------------------------------------------------------------------
#define EDGES_PER_WAVE 4

__global__ void __launch_bounds__(256)
gin_scatter_add(const float* __restrict__ h,
                const int* __restrict__ src,
                const int* __restrict__ dst,
                float* __restrict__ agg) {
    const int wv   = __builtin_amdgcn_readfirstlane(threadIdx.x >> 5);
    const int lane = threadIdx.x & 31;
    const int e0   = (blockIdx.x * 8 + wv) * EDGES_PER_WAVE;

#pragma unroll
    for (int j = 0; j < EDGES_PER_WAVE; ++j) {
        const int e = e0 + j;
        if (e < N_EDGES) {
            const int u = src[e];                 // uniform -> s_load
            const int v = dst[e];                 // uniform -> s_load

            // 32 lanes x float4 = full 512 B row
            float4 val = ((const float4*)h)[u * 32 + lane];

            float* arow = agg + v * KDIM + lane * 4;   // 32-bit offset, saddr form
            atomicAdd(arow + 0, val.x);
            atomicAdd(arow + 1, val.y);
            atomicAdd(arow + 2, val.z);
            atomicAdd(arow + 3, val.w);
        }
    }
}

// ---------------------------------------------------------------------------
// out[M,NOUT] = X[M,128] @ W[128,NOUT] + bias   via V_WMMA_F32_16X16X4_F32.
// One block per 32-row M slab (100000 = 32 * 3125, exact). Wave w owns
// N tile [16w,16w+16) and computes two stacked 16x16 output tiles, reusing
// its B fragment across both. LDS pitch 132 -> bank = (4*row + k) % 64,
// conflict-free across the 16 lanes of each half.
// ---------------------------------------------------------------------------
template <int NOUT>
__global__ void __launch_bounds__(256)
gin_gemm_wmma(const float* __restrict__ X,
              const float* __restrict__ W,
              const float* __restrict__ bias,
              float* __restrict__ out) {
    __shared__ float xs[32 * 132];

    const int m_base = blockIdx.x * 32;

    // Cooperative load of the 32x128 X slab (1024 float4, 4 per thread).
    for (int idx = threadIdx.x; idx < 32 * 32; idx += 256) {
        int r  = idx >> 5;
        int c4 = idx & 31;
        float4 v = ((const float4*)(X + (m_base + r) * KDIM))[c4];
        float* s = &xs[r * 132 + c4 * 4];
        s[0] = v.x; s[1] = v.y; s[2] = v.z; s[3] = v.w;
    }
    __syncthreads();

    const int wave   = threadIdx.x >> 5;
    const int lane   = threadIdx.x & 31;
    const int n_base = wave * 16;
    if (n_base >= NOUT) return;               // uniform per wave (NOUT=64 case)

    const int half = lane >> 4;               // 0: K pair {0,1}; 1: K pair {2,3}
    const int rl   = lane & 15;               // A row / B column within tile

    v8f acc0 = {0.f, 0.f, 0.f, 0.f, 0.f, 0.f, 0.f, 0.f};
    v8f acc1 = {0.f, 0.f, 0.f, 0.f, 0.f, 0.f, 0.f, 0.f};

    for (int k0 = 0; k0 < KDIM; k0 += 4) {
        const int k = k0 + 2 * half;
        v2f b, a0, a1;
        b.x  = W[k * NOUT + n_base + rl];               // const stride: shifts + saddr
        b.y  = W[(k + 1) * NOUT + n_base + rl];
        a0.x = xs[rl * 132 + k];
        a0.y = xs[rl * 132 + k + 1];
        a1.x = xs[(16 + rl) * 132 + k];
        a1.y = xs[(16 + rl) * 132 + k + 1];
        // D = A(16x4) * B(4x16) + C
        acc0 = __builtin_amdgcn_wmma_f32_16x16x4_f32(false, a0, false, b,
                                                     (short)0, acc0, false, false);
        acc1 = __builtin_amdgcn_wmma_f32_16x16x4_f32(false, a1, false, b,
                                                     (short)0, acc1, false, false);
    }

    const int   col  = n_base + rl;
    const float bcol = bias[col];
#pragma unroll
    for (int i = 0; i < 8; ++i) {
        // C/D layout: VGPR i -> M = i + 8*half, N = lane&15
        out[(m_base +      i + 8 * half) * NOUT + col] = acc0[i] + bcol;
        out[(m_base + 16 + i + 8 * half) * NOUT + col] = acc1[i] + bcol;
    }
}

// ---------------------------------------------------------------------------
extern "C" void kernel_launch(void* const* d_in, const int* in_sizes, int n_in,
                              void* d_out, int out_size, void* d_ws, size_t ws_size,
                              hipStream_t stream) {
    const float* h0  = (const float*)d_in[0];
    const int*   ei  = (const int*)d_in[1];
    const int*   src = ei;             // edge_index[0]
    const int*   dst = ei + N_EDGES;   // edge_index[1]

    const float* Wv[5];
    const float* bv[5];
    for (int i = 0; i < 5; ++i) {
        Wv[i] = (const float*)d_in[2 + 2 * i];
        bv[i] = (const float*)d_in[3 + 2 * i];
    }

    float* bufA = (float*)d_ws;
    float* bufB = bufA + (size_t)N_NODES * KDIM;
    float* agg  = bufB + (size_t)N_NODES * KDIM;

    const int n4 = N_NODES * KDIM / 4;
    dim3 cpGrid((n4 + 255) / 256);
    // 8 waves/block * 4 edges/wave = 32 edges per block; 3.2M / 32 = 100000 blocks
    dim3 scGrid(N_EDGES / (8 * EDGES_PER_WAVE));
    dim3 gmGrid(N_NODES / 32);                               // 3125 slabs, exact

    const float* cur = h0;
    for (int layer = 0; layer < 5; ++layer) {
        float* outp = (layer == 4) ? (float*)d_out
                                   : ((layer & 1) ? bufB : bufA);

        gin_copy_f4    <<<cpGrid, 256, 0, stream>>>((float4*)agg, (const float4*)cur, n4);
        gin_scatter_add<<<scGrid, 256, 0, stream>>>(cur, src, dst, agg);
        if (layer == 4)
            gin_gemm_wmma<64> <<<gmGrid, 256, 0, stream>>>(agg, Wv[layer], bv[layer], outp);
        else
            gin_gemm_wmma<128><<<gmGrid, 256, 0, stream>>>(agg, Wv[layer], bv[layer], outp);

        cur = outp;
    }
}